// Weight_MS_44573170598310
// MI455X (gfx1250) — compile-verified
//
#include <hip/hip_runtime.h>
#include <math.h>

namespace {

constexpr int   N_    = 4096;
constexpr int   D_    = 128;
constexpr int   ROWS  = 16;    // anchors per workgroup
constexpr int   TPB   = 512;   // 16 waves (wave32)
constexpr float ALPHA = 10.0f;
constexpr float BETA  = 2.0f;
constexpr float BASE  = 0.5f;

typedef float v2f __attribute__((ext_vector_type(2)));
typedef float v8f __attribute__((ext_vector_type(8)));

// f32 -> order-preserving u32 (ascending)
__device__ __forceinline__ unsigned sortKey(float f) {
  unsigned u = __float_as_uint(f);
  return (u & 0x80000000u) ? ~u : (u | 0x80000000u);
}
__device__ __forceinline__ float keyVal(unsigned long long k) {
  unsigned u = (unsigned)(k >> 12);
  return (u & 0x80000000u) ? __uint_as_float(u & 0x7fffffffu)
                           : __uint_as_float(~u);
}

// dynamic LDS layout (bytes):
//   float  Alds [ROWS*D_]            16*128*4   =   8192
//   float  strip[ROWS*N_]            16*4096*4  = 262144
//   u64    keys [N_]                 4096*8     =  32768
//   u64    posKeys[8]                            =     64
//   float  redf[TPB]                              =  2048
//   int    redi[TPB]                              =  2048
constexpr size_t MINE_SMEM =
    (size_t)(ROWS * D_ + ROWS * N_) * 4 + (size_t)(N_ + 8) * 8 + (size_t)TPB * 8;

// ---------------------------------------------------------------- K0: scan
__global__ void scan_init_kernel(const int* __restrict__ ap,
                                 const int* __restrict__ an,
                                 int* __restrict__ aOff, int* __restrict__ bOff,
                                 int* __restrict__ cAcc) {
  __shared__ int tot[N_];
  int tid = threadIdx.x;
  for (int i = tid; i < N_; i += blockDim.x) tot[i] = ap[i] + an[i];
  __syncthreads();
  if (tid == 0) {
    int run = 0;
    for (int i = 0; i < N_; ++i) { int t = tot[i]; tot[i] = run; run += t; }
    *cAcc = 0;
  }
  __syncthreads();
  for (int i = tid; i < N_; i += blockDim.x) {
    aOff[i] = tot[i];
    bOff[i] = tot[i] + ap[i];
  }
}

// ------------------------------------------------- K1: WMMA GEMM + mining
__global__ __launch_bounds__(TPB) void ms_mine_kernel(
    const float* __restrict__ X, const float* __restrict__ margin,
    const float* __restrict__ weight, const int* __restrict__ targets,
    const int* __restrict__ ap, const int* __restrict__ an,
    const int* __restrict__ aOff, const int* __restrict__ bOff,
    float* __restrict__ lossArr, int* __restrict__ cAcc,
    float* __restrict__ out) {
  extern __shared__ unsigned char smem_raw[];
  float* Alds  = (float*)smem_raw;                         // ROWS*D_
  float* strip = Alds + ROWS * D_;                         // ROWS*N_
  unsigned long long* keys    = (unsigned long long*)(strip + ROWS * N_);
  unsigned long long* posKeys = keys + N_;                 // 8 entries
  float* redf = (float*)(posKeys + 8);                     // TPB
  int*   redi = (int*)(redf + TPB);                        // TPB
  __shared__ float sPmin;
  __shared__ int   sPosCnt;

  const int tid  = threadIdx.x;
  const int lane = tid & 31;
  const int wave = tid >> 5;
  const int rowBase = blockIdx.x * ROWS;

  // ---- Phase A: stage 16 anchor rows of X into LDS.
  // gfx1250 async global->LDS DMA path (ASYNCcnt-tracked, no VGPR
  // round-trip). The builtin's first param is a global (AS1) int4*.
#if __has_builtin(__builtin_amdgcn_global_load_async_to_lds_b128)
  {
    typedef int v4i_ __attribute__((vector_size(16)));
    typedef __attribute__((address_space(1))) v4i_* gp_t;
    typedef __attribute__((address_space(3))) v4i_* lp_t;
    // ROWS*D_*4 = 8192 B = 512 x b128 : one async b128 per thread
    gp_t g = (gp_t)(const_cast<float*>(X) + (size_t)rowBase * D_) + tid;
    lp_t l = (lp_t)Alds + tid;
    __builtin_amdgcn_global_load_async_to_lds_b128(g, l, 0, 0);
#if __has_builtin(__builtin_amdgcn_s_wait_asynccnt)
    __builtin_amdgcn_s_wait_asynccnt(0);
#else
    asm volatile("s_wait_asynccnt 0x0" ::: "memory");
#endif
  }
#else
  for (int e = tid; e < ROWS * D_; e += TPB)
    Alds[e] = X[rowBase * D_ + e];
#endif
  __syncthreads();

  // ---- Phase B: sim strip = A (16xD) * X^T  via v_wmma_f32_16x16x4_f32
  // A-frag (16x4 f32): lanes 0-15 -> M=lane, K=0,1 ; lanes 16-31 -> K=2,3
  // B-frag (4x16 f32): VGPR j, lanes 0-15 -> row j ; lanes 16-31 -> row j+2
  // C/D (16x16 f32):   VGPR v, lanes 0-15 -> M=v   ; lanes 16-31 -> M=v+8
  {
    const int hi = lane >> 4;
    const int mr = lane & 15;

    // Preload this wave's A fragments ONCE into registers (64 VGPRs):
    // reused across all 16 column-block iterations.
    v2f afrag[D_ / 4];
    #pragma unroll
    for (int kk = 0; kk < D_ / 4; ++kk) {
      const int ko = 4 * kk + 2 * hi;
      afrag[kk].x = Alds[mr * D_ + ko];
      afrag[kk].y = Alds[mr * D_ + ko + 1];
    }

    for (int cb = wave * 16; cb < N_; cb += (TPB / 32) * 16) {
      v8f acc = {0.f, 0.f, 0.f, 0.f, 0.f, 0.f, 0.f, 0.f};
      const float* xcol = X + (size_t)(cb + mr) * D_;   // L2-resident
      #pragma unroll
      for (int kk = 0; kk < D_ / 4; ++kk) {
        v2f b = *(const v2f*)(xcol + 4 * kk + 2 * hi);  // 8B aligned
        acc = __builtin_amdgcn_wmma_f32_16x16x4_f32(
            false, afrag[kk], false, b, (short)0, acc, false, false);
      }
      #pragma unroll
      for (int v = 0; v < 8; ++v) {
        const int row = v + 8 * hi;
        strip[row * N_ + cb + mr] = acc[v];
      }
    }
  }
  __syncthreads();

  // ---- Phase C: per-anchor mining from the LDS strip
  for (int r = 0; r < ROWS; ++r) {
    const int   i    = rowBase + r;
    const int   tgt  = targets[i];
    const float marg = margin[i];
    if (tid == 0) sPosCnt = 0;
    __syncthreads();

    // pass 1: build sortable keys, gather positives, neg stats
    float lmax = -__builtin_inff();
    float lsum = 0.f;
    int   lcnt = 0;
    const float* srow = strip + r * N_;
    for (int j = tid; j < N_; j += TPB) {
      const float v = srow[j];
      if (targets[j] == tgt) {
        if (v < 1.0f) {
          int idx = atomicAdd(&sPosCnt, 1);
          if (idx < 8)
            posKeys[idx] = ((unsigned long long)sortKey(v) << 12) | (unsigned)j;
        }
        keys[j] = ~0ull;  // padding: sorts past all finite negatives
      } else {
        keys[j] = ((unsigned long long)sortKey(v) << 12) | (unsigned)j;
        lmax = fmaxf(lmax, v);
        lsum += v;
        lcnt++;
      }
    }
    __syncthreads();

    // reduce: nmax, negCnt
    redf[tid] = lmax; redi[tid] = lcnt;
    __syncthreads();
    for (int k = TPB / 2; k > 0; k >>= 1) {
      if (tid < k) { redf[tid] = fmaxf(redf[tid], redf[tid + k]); redi[tid] += redi[tid + k]; }
      __syncthreads();
    }
    const float nmax   = redf[0];
    const int   negCnt = redi[0];
    __syncthreads();
    // reduce: negSum (for last-anchor mean)
    redf[tid] = lsum;
    __syncthreads();
    for (int k = TPB / 2; k > 0; k >>= 1) {
      if (tid < k) redf[tid] += redf[tid + k];
      __syncthreads();
    }
    const float negSum = redf[0];
    __syncthreads();

    // sort the <=8 positives, pmin
    if (tid == 0) {
      int pc = sPosCnt; if (pc > 8) pc = 8;
      for (int a2 = 1; a2 < pc; ++a2) {
        unsigned long long kv = posKeys[a2];
        int b2 = a2 - 1;
        while (b2 >= 0 && posKeys[b2] > kv) { posKeys[b2 + 1] = posKeys[b2]; --b2; }
        posKeys[b2 + 1] = kv;
      }
      sPmin = (pc > 0) ? keyVal(posKeys[0]) : __builtin_inff();
    }
    __syncthreads();
    const float pmin = sPmin;

    // bitonic sort: 4096 u64 (value, col) keys, ascending == jnp.sort stable
    for (int k = 2; k <= N_; k <<= 1) {
      for (int j = k >> 1; j > 0; j >>= 1) {
        for (int idx = tid; idx < N_; idx += TPB) {
          const int ixj = idx ^ j;
          if (ixj > idx) {
            unsigned long long u = keys[idx], w = keys[ixj];
            const bool up = ((idx & k) == 0);
            if ((u > w) == up) { keys[idx] = w; keys[ixj] = u; }
          }
        }
        __syncthreads();
      }
    }

    // survivors are a suffix: first kept sorted position
    int lmin = N_;
    for (int s = tid; s < negCnt; s += TPB) {
      const float v = keyVal(keys[s]);
      if (v + marg - pmin > 0.f) lmin = min(lmin, s);
    }
    redi[tid] = lmin;
    __syncthreads();
    for (int k = TPB / 2; k > 0; k >>= 1) {
      if (tid < k) redi[tid] = min(redi[tid], redi[tid + k]);
      __syncthreads();
    }
    const int firstKeep = redi[0];
    __syncthreads();

    // neg exp sum with rank-indexed weights (coalesced weight reads)
    const int bO = bOff[i];
    float lexp = 0.f;
    for (int s = tid; s < negCnt; s += TPB) {
      if (s >= firstKeep) {
        const float v = keyVal(keys[s]);
        const float w = weight[bO + (s - firstKeep)];
        lexp += expf(ALPHA * (v * w - BASE));
      }
    }
    redf[tid] = lexp;
    __syncthreads();
    for (int k = TPB / 2; k > 0; k >>= 1) {
      if (tid < k) redf[tid] += redf[tid + k];
      __syncthreads();
    }

    if (tid == 0) {
      const float negExp = redf[0];
      const bool  anyNeg = firstKeep < negCnt;
      int pc = sPosCnt; if (pc > 8) pc = 8;
      const int aO = aOff[i];
      float posExp = 0.f, pvSum = 0.f;
      int   rank = 0;
      bool  anyPos = false;
      for (int q = 0; q < pc; ++q) {
        const float v = keyVal(posKeys[q]);
        pvSum += v;
        if (nmax - v + marg > 0.f) {
          const float w = weight[aO + rank];
          posExp += expf(-BETA * (v * w - BASE));
          rank++;
          anyPos = true;
        }
      }
      const bool  has    = anyPos && anyNeg;
      const float loss_i = (2.0f / BETA) * log1pf(posExp) +
                           (2.0f / ALPHA) * log1pf(negExp);
      lossArr[i] = has ? loss_i : 0.f;
      if (has) atomicAdd(cAcc, ap[i] + an[i]);
      if (i == N_ - 1) {
        out[2] = pvSum / (float)pc;            // mean_pos_sim (last anchor)
        out[3] = negSum / (float)negCnt;       // mean_neg_sim (last anchor)
      }
    }
    __syncthreads();
  }
}

// ------------------------------------------------------------- K2: finalize
__global__ void finalize_kernel(const float* __restrict__ lossArr,
                                const int* __restrict__ cAcc,
                                float* __restrict__ out) {
  __shared__ float red[256];
  int tid = threadIdx.x;
  float s = 0.f;
  for (int i = tid; i < N_; i += 256) s += lossArr[i];
  red[tid] = s;
  __syncthreads();
  for (int k = 128; k > 0; k >>= 1) {
    if (tid < k) red[tid] += red[tid + k];
    __syncthreads();
  }
  if (tid == 0) {
    out[0] = red[0] / (float)N_;
    out[1] = (float)(*cAcc) / (float)N_;
  }
}

}  // namespace

extern "C" void kernel_launch(void* const* d_in, const int* in_sizes, int n_in,
                              void* d_out, int out_size, void* d_ws, size_t ws_size,
                              hipStream_t stream) {
  const float* X       = (const float*)d_in[0];
  const float* margin  = (const float*)d_in[1];
  const float* weight  = (const float*)d_in[2];
  const int*   targets = (const int*)d_in[3];
  const int*   ap      = (const int*)d_in[4];
  const int*   an      = (const int*)d_in[5];
  float*       out     = (float*)d_out;

  int*   aOff    = (int*)d_ws;
  int*   bOff    = aOff + N_;
  float* lossArr = (float*)(bOff + N_);
  int*   cAcc    = (int*)(lossArr + N_);

  (void)hipFuncSetAttribute(reinterpret_cast<const void*>(&ms_mine_kernel),
                            hipFuncAttributeMaxDynamicSharedMemorySize,
                            (int)MINE_SMEM);

  scan_init_kernel<<<1, 256, 0, stream>>>(ap, an, aOff, bOff, cAcc);
  ms_mine_kernel<<<N_ / ROWS, TPB, MINE_SMEM, stream>>>(
      X, margin, weight, targets, ap, an, aOff, bOff, lossArr, cAcc, out);
  finalize_kernel<<<1, 256, 0, stream>>>(lossArr, cAcc, out);
}